// GELU268_23648089932069
// MI455X (gfx1250) — compile-verified
//
#include <hip/hip_runtime.h>
#include <math.h>

// Problem sizes (fixed by the reference setup)
#define D     1024
#define D4    256       // D / 4 (float4 granularity)
#define BT    16384     // B*T = 4*4096
#define NROW  512       // buffer rows
#define NBLK1 256       // blocks in kernel 1
#define WAVES_PER_BLK 8
#define TOK_PER_WAVE  8 // 16384 tokens / (256 blocks * 8 waves)

// Workspace layout (float units):
//  [0, BT)                : gate per token              (64 KB)
//  [BT, BT+NBLK1*D)       : per-block y1 partial sums   (1 MB)
//  [SCAL_OFF, +8)         : boost, damp, valid, idx(bits)
#define GATE_OFF 0
#define PART_OFF BT
#define SCAL_OFF (BT + NBLK1 * D)

typedef __attribute__((ext_vector_type(2))) float v2f;
typedef __attribute__((ext_vector_type(8))) float v8f;

#define PER_COMP(OP) OP(x) OP(y) OP(z) OP(w)

// Hardware tanh (V_TANH_F32, single TRANS32 op) instead of libm polynomial.
__device__ __forceinline__ float fast_tanh(float v) {
#if __has_builtin(__builtin_amdgcn_tanhf)
    return __builtin_amdgcn_tanhf(v);
#else
    float r;
    // Inline asm bypasses compiler hazard handling: ISA requires 1 independent
    // op or V_NOP after a TRANS op before its result is consumed.
    asm volatile("v_tanh_f32 %0, %1\n\tv_nop" : "=v"(r) : "v"(v));
    return r;
#endif
}

__device__ __forceinline__ float gelu_f(float v) {
    float v3 = v * v * v;
    return 0.5f * v * (1.0f + fast_tanh(0.7978845608028654f * (v + 0.044715f * v3)));
}

// Full-wave (wave32) sum; butterfly leaves result in every lane.
__device__ __forceinline__ float wave_sum(float v) {
#pragma unroll
    for (int m = 16; m >= 1; m >>= 1) v += __shfl_xor(v, m, 32);
    return v;
}

// ---------------------------------------------------------------------------
// Kernel 1: per-token gate + deterministic per-block partial sums of y1.
// One wave per token; all global traffic is float4 (b128).
// ---------------------------------------------------------------------------
__global__ __launch_bounds__(256)
void k1_gate_partials(const float* __restrict__ xf,
                      const float* __restrict__ p_lti, const float* __restrict__ p_lto,
                      const float* __restrict__ p_lai, const float* __restrict__ p_lao,
                      const float* __restrict__ ema_x, const float* __restrict__ ema_x2,
                      const float* __restrict__ ema_y, const float* __restrict__ ema_y2,
                      float* __restrict__ ws) {
    __shared__ float4 s_mx[D4], s_isx[D4], s_my[D4], s_isy[D4];
    __shared__ float4 s_acc[WAVES_PER_BLK][D4];

    const int tid  = threadIdx.x;
    const int lane = tid & 31;
    const int wid  = tid >> 5;

    // Stage detached z-statistics once per block (exactly one float4 per thread).
    {
        const int d4 = tid;   // D4 == blockDim.x == 256
        float4 mx  = ((const float4*)ema_x)[d4];
        float4 mx2 = ((const float4*)ema_x2)[d4];
        float4 my  = ((const float4*)ema_y)[d4];
        float4 my2 = ((const float4*)ema_y2)[d4];
        float4 isx, isy;
#define STAT(c) \
        isx.c = 1.0f / (sqrtf(fmaxf(mx2.c - mx.c * mx.c, 0.0f)) + 1e-5f); \
        isy.c = 1.0f / (sqrtf(fmaxf(my2.c - my.c * my.c, 0.0f)) + 1e-5f);
        PER_COMP(STAT)
#undef STAT
        s_mx[d4] = mx; s_my[d4] = my; s_isx[d4] = isx; s_isy[d4] = isy;
    }
    __syncthreads();

    const float tau_in  = expf(p_lti[0]);
    const float tau_out = expf(p_lto[0]);
    const float a_in    = 1.0f / (1.0f + expf(-p_lai[0]));
    const float a_out   = 1.0f / (1.0f + expf(-p_lao[0]));

    float4 acc[8];
#pragma unroll
    for (int j = 0; j < 8; ++j) acc[j] = make_float4(0.f, 0.f, 0.f, 0.f);

    const int waveGlobal = blockIdx.x * WAVES_PER_BLK + wid;
    const float4* x4 = (const float4*)xf;
#pragma unroll 1
    for (int t = 0; t < TOK_PER_WAVE; ++t) {
        const int token = waveGlobal * TOK_PER_WAVE + t;
        const float4* xr = x4 + (size_t)token * D4;
        if (t + 1 < TOK_PER_WAVE)
            __builtin_prefetch(xr + D4 + lane * 8, 0, 0);  // next token row

        float4 yv[8];
        float s_in = 0.0f, s_out = 0.0f;
#pragma unroll
        for (int j = 0; j < 8; ++j) {
            const int d4 = j * 32 + lane;
            float4 xv  = xr[d4];
            float4 mx  = s_mx[d4],  isx = s_isx[d4];
            float4 my  = s_my[d4],  isy = s_isy[d4];
            float4 y;
#define ELEM(c) { \
            float yc = gelu_f(xv.c); y.c = yc; \
            float zi = (xv.c - mx.c) * isx.c;  s_in  += zi * zi; \
            float zo = (yc   - my.c) * isy.c;  s_out += zo * zo; }
            PER_COMP(ELEM)
#undef ELEM
            yv[j] = y;
        }
        s_in  = wave_sum(s_in)  * (1.0f / D);
        s_out = wave_sum(s_out) * (1.0f / D);
        float g_in  = 1.0f - a_in  + a_in  * expf(-tau_in  * s_in);
        float g_out = 1.0f - a_out + a_out * expf(-tau_out * s_out);
        float gate  = g_in * g_out;
        if (lane == 0) ws[GATE_OFF + token] = gate;
#pragma unroll
        for (int j = 0; j < 8; ++j) {
#define ACC(c) acc[j].c += yv[j].c * gate;
            PER_COMP(ACC)
#undef ACC
        }
    }

    // Wave-private slots, then fixed-order cross-wave reduction (deterministic).
#pragma unroll
    for (int j = 0; j < 8; ++j) s_acc[wid][j * 32 + lane] = acc[j];
    __syncthreads();
    {
        const int d4 = tid;
        float4 s = make_float4(0.f, 0.f, 0.f, 0.f);
#pragma unroll
        for (int w = 0; w < WAVES_PER_BLK; ++w) {
            float4 t = s_acc[w][d4];
#define ADD(c) s.c += t.c;
            PER_COMP(ADD)
#undef ADD
        }
        ((float4*)(ws + PART_OFF))[(size_t)blockIdx.x * D4 + d4] = s;
    }
}

// ---------------------------------------------------------------------------
// Kernel 2: y1_mean -> m_n, row norms, sims via V_WMMA_F32_16X16X4_F32,
// argmax, scalar outputs. Single block of 8 waves.
// ---------------------------------------------------------------------------
__global__ __launch_bounds__(256)
void k2_sims_argmax(const float* __restrict__ buf, const float* __restrict__ facil,
                    const unsigned char* __restrict__ mask,
                    const float* __restrict__ p_lkb, const float* __restrict__ p_lkd,
                    float* __restrict__ ws) {
    __shared__ float4 s_m4[D4];       // y1_mean, then m_n (aliased as float[])
    __shared__ float  s_n2[NROW];     // |buf_row|^2
    __shared__ float  s_sims[NROW];
    __shared__ float  s_red[WAVES_PER_BLK];
    __shared__ float  s_invn;

    const int tid = threadIdx.x, lane = tid & 31, wid = tid >> 5;
    float* s_m = reinterpret_cast<float*>(s_m4);

    // 1) reduce per-block partials -> y1_mean (fixed order, float4 loads).
    float ps;
    {
        const int d4 = tid;   // exactly one float4 per thread
        const float4* part4 = (const float4*)(ws + PART_OFF);
        float4 s = make_float4(0.f, 0.f, 0.f, 0.f);
        for (int b = 0; b < NBLK1; ++b) {
            float4 t = part4[(size_t)b * D4 + d4];
#define ADD(c) s.c += t.c;
            PER_COMP(ADD)
#undef ADD
        }
#define SCALE(c) s.c *= (1.0f / BT);
        PER_COMP(SCALE)
#undef SCALE
        s_m4[d4] = s;
        ps = s.x * s.x + s.y * s.y + s.z * s.z + s.w * s.w;
    }
    __syncthreads();

    // 2) |y1_mean| and normalize -> m_n
    ps = wave_sum(ps);
    if (lane == 0) s_red[wid] = ps;
    __syncthreads();
    if (tid == 0) {
        float t = 0.0f;
        for (int w = 0; w < WAVES_PER_BLK; ++w) t += s_red[w];
        s_invn = 1.0f / fmaxf(sqrtf(t), 1e-12f);
    }
    __syncthreads();
    {
        const float invn = s_invn;
        float4 m = s_m4[tid];
#define SCL(c) m.c *= invn;
        PER_COMP(SCL)
#undef SCL
        s_m4[tid] = m;
    }
    __syncthreads();

    // 3) buf row squared norms (float4 loads).
    const float4* buf4 = (const float4*)buf;
    for (int r = wid; r < NROW; r += WAVES_PER_BLK) {
        float p = 0.0f;
#pragma unroll
        for (int j = 0; j < 8; ++j) {
            float4 v = buf4[(size_t)r * D4 + j * 32 + lane];
            p += v.x * v.x + v.y * v.y + v.z * v.z + v.w * v.w;
        }
        p = wave_sum(p);
        if (lane == 0) s_n2[r] = p;
    }
    __syncthreads();

    // 4) sims = buf_tile(16xD) . m_n  via FP32 WMMA 16x16x4, K accumulated in steps of 4.
    //    A layout (32-bit 16x4): lanes 0-15 -> K=0,1 ; lanes 16-31 -> K=2,3 ; row = r0+(lane&15).
    //    B replicates the m_n K-chunk across all 16 columns -> every C column holds the dot.
    for (int tile = wid; tile < NROW / 16; tile += WAVES_PER_BLK) {
        const int r0   = tile * 16;
        const int row  = r0 + (lane & 15);
        const int ksub = (lane < 16) ? 0 : 2;
        const float* ap = buf + (size_t)row * D + ksub;
        v8f c = {0.f, 0.f, 0.f, 0.f, 0.f, 0.f, 0.f, 0.f};
#pragma unroll 4
        for (int k = 0; k < D; k += 4) {
            v2f a, b;
            a.x = ap[k];            a.y = ap[k + 1];
            b.x = s_m[k + ksub];    b.y = s_m[k + ksub + 1];
            c = __builtin_amdgcn_wmma_f32_16x16x4_f32(
                    false, a, false, b, (short)0, c, false, false);
        }
        // C layout: VGPR j -> M=j (lanes 0-15, N=lane) / M=8+j (lanes 16-31, N=lane-16).
        if (lane == 0 || lane == 16) {
#pragma unroll
            for (int j = 0; j < 8; ++j) {
                const int r = r0 + ((lane == 0) ? j : (8 + j));
                float raw = c[j];
                float sim = raw / fmaxf(sqrtf(s_n2[r]), 1e-12f);
                s_sims[r] = (mask[r] != 0) ? sim : -1.0f;
            }
        }
    }
    __syncthreads();

    // 5) argmax (first-index tie-break) + scalar outputs, by wave 0.
    if (wid == 0) {
        float best = -3.0e38f; int bidx = 0;
        for (int i = 0; i < NROW / 32; ++i) {
            const int r = i * 32 + lane;           // lane-local indices ascend
            float v = s_sims[r];
            if (v > best) { best = v; bidx = r; }  // strict '>' keeps lowest index
        }
#pragma unroll
        for (int m = 16; m >= 1; m >>= 1) {
            float ov = __shfl_xor(best, m, 32);
            int   oi = __shfl_xor(bidx, m, 32);
            if (ov > best || (ov == best && oi < bidx)) { best = ov; bidx = oi; }
        }
        // validity: all finite & |v| >= 1e-6
        int ok = 1;
#pragma unroll
        for (int j = 0; j < 8; ++j) {
            float4 vv = buf4[(size_t)bidx * D4 + j * 32 + lane];
#define CHK(c) if (!__builtin_isfinite(vv.c)) ok = 0;
            PER_COMP(CHK)
#undef CHK
        }
#pragma unroll
        for (int m = 16; m >= 1; m >>= 1) ok &= __shfl_xor(ok, m, 32);

        if (lane == 0) {
            float sim_val = fminf(fmaxf(best, 0.0f), 1.0f);
            int fired = sim_val > 0.88f;
            float facil_level = facil[bidx] * (fired ? 2.0f : 1.0f);
            float k_boost = fminf(fmaxf(expf(p_lkb[0]), 0.01f), 4.0f);
            float k_damp  = fminf(fmaxf(expf(p_lkd[0]), 0.01f), 0.9f);
            float mod   = (facil_level - 1.0f) * sim_val;
            float boost = 1.0f + k_boost * mod;
            float damp  = fmaxf(0.01f, 1.0f - k_damp * mod);
            float nv    = sqrtf(s_n2[bidx]);
            int valid   = ok && __builtin_isfinite(nv) && (nv >= 1e-6f);
            float* sw = ws + SCAL_OFF;
            sw[0] = boost;
            sw[1] = damp;
            sw[2] = valid ? 1.0f : 0.0f;
            ((int*)sw)[3] = bidx;
        }
    }
}

// ---------------------------------------------------------------------------
// Kernel 3: per-token projection + blend. One wave per token; y1 lives in
// VGPRs; x re-read hits L2 (192 MB holds the tensor); all traffic b128.
// ---------------------------------------------------------------------------
__global__ __launch_bounds__(256)
void k3_project_out(const float* __restrict__ xf, const float* __restrict__ buf,
                    const float* __restrict__ ws, float* __restrict__ outf) {
    const int tid = threadIdx.x, lane = tid & 31, wid = tid >> 5;
    const int token = blockIdx.x * WAVES_PER_BLK + wid;

    const float* sw   = ws + SCAL_OFF;
    const float boost = sw[0];
    const float damp  = sw[1];
    const bool  valid = sw[2] != 0.0f;
    const int   idx   = ((const int*)sw)[3];
    const float4* v4  = (const float4*)buf + (size_t)idx * D4;

    const float gate = ws[GATE_OFF + token];
    const float4* xr = (const float4*)xf + (size_t)token * D4;

    float4 y1[8], vv[8];
    float p = 0.0f;
#pragma unroll
    for (int j = 0; j < 8; ++j) {
        const int d4 = j * 32 + lane;
        float4 xv = xr[d4];
        float4 w  = v4[d4];
        float4 y;
#define ELEM(c) { float t = gelu_f(xv.c) * gate; y.c = t; p += t * w.c; }
        PER_COMP(ELEM)
#undef ELEM
        y1[j] = y; vv[j] = w;
    }
    p = wave_sum(p);   // proj, in every lane

    float4* orow = (float4*)outf + (size_t)token * D4;
#pragma unroll
    for (int j = 0; j < 8; ++j) {
        const int d4 = j * 32 + lane;
        float4 o;
#define ELEM(c) { float yp = p * vv[j].c; \
                  o.c = valid ? (yp * boost + (y1[j].c - yp) * damp) : y1[j].c; }
        PER_COMP(ELEM)
#undef ELEM
        orow[d4] = o;
    }
}

// ---------------------------------------------------------------------------
extern "C" void kernel_launch(void* const* d_in, const int* in_sizes, int n_in,
                              void* d_out, int out_size, void* d_ws, size_t ws_size,
                              hipStream_t stream) {
    const float* x      = (const float*)d_in[0];
    const float* lti    = (const float*)d_in[1];
    const float* lto    = (const float*)d_in[2];
    const float* lai    = (const float*)d_in[3];
    const float* lao    = (const float*)d_in[4];
    const float* lkb    = (const float*)d_in[5];
    const float* lkd    = (const float*)d_in[6];
    const float* ema_x  = (const float*)d_in[7];
    const float* ema_x2 = (const float*)d_in[8];
    const float* ema_y  = (const float*)d_in[9];
    const float* ema_y2 = (const float*)d_in[10];
    const float* buf    = (const float*)d_in[11];
    const float* facil  = (const float*)d_in[12];
    const unsigned char* mask = (const unsigned char*)d_in[13];
    float* ws  = (float*)d_ws;
    float* out = (float*)d_out;

    // ws needs (BT + NBLK1*D + 8) floats ~= 1.06 MB.
    k1_gate_partials<<<dim3(NBLK1), dim3(256), 0, stream>>>(
        x, lti, lto, lai, lao, ema_x, ema_x2, ema_y, ema_y2, ws);
    k2_sims_argmax<<<dim3(1), dim3(256), 0, stream>>>(
        buf, facil, mask, lkb, lkd, ws);
    k3_project_out<<<dim3(BT / WAVES_PER_BLK), dim3(256), 0, stream>>>(
        x, buf, ws, out);
}